// SAGPool_55233279426756
// MI455X (gfx1250) — compile-verified
//
#include <hip/hip_runtime.h>
#include <math.h>

typedef __attribute__((ext_vector_type(2))) float v2f;
typedef __attribute__((ext_vector_type(8))) float v8f;

#define NTOT 65536   // total nodes
#define DIM  256     // features
#define NE   2097152 // edges
#define NB   128     // graphs
#define NN   512     // nodes per graph
#define KK   256     // top-k per graph

// ---- init: deg = 1 (self loop weight), mask = 0 --------------------------
__global__ void k_init(float* __restrict__ deg, float* __restrict__ mask) {
    int i = blockIdx.x * blockDim.x + threadIdx.x;
    if (i < NTOT) { deg[i] = 1.0f; mask[i] = 0.0f; }
}

// ---- degree accumulation over edges (segment_sum of ew by col) -----------
__global__ void k_deg(const int* __restrict__ ei, const float* __restrict__ ew,
                      float* __restrict__ deg) {
    int e = blockIdx.x * blockDim.x + threadIdx.x;
    if (e < NE) {
        int c = ei[NE + e];
        atomicAdd(&deg[c], ew[e]);
    }
}

// ---- h = x @ W via V_WMMA_F32_16X16X4_F32 (full f32 precision) -----------
// One wave32 handles 16 rows. A (16x4 f32): lanes 0-15 hold K=0,1; lanes
// 16-31 hold K=2,3 (ISA 7.12.2). B (4x16): W[k] replicated across all 16
// columns, same K-per-lane split as A. Every column of D = dot(x_row, W).
__global__ void k_h_wmma(const float* __restrict__ x, const float* __restrict__ W,
                         float* __restrict__ h) {
    int wave = (blockIdx.x * blockDim.x + threadIdx.x) >> 5;
    int lane = threadIdx.x & 31;
    int row_base = wave * 16;
    int m = lane & 15;
    int khalf = (lane >> 4) << 1;             // 0 for lanes 0-15, 2 for 16-31
    const float* xrow = x + (size_t)(row_base + m) * DIM;

    v8f acc = {0.f,0.f,0.f,0.f,0.f,0.f,0.f,0.f};
    for (int k0 = 0; k0 < DIM; k0 += 4) {
        v2f a;  a.x  = xrow[k0 + khalf]; a.y  = xrow[k0 + khalf + 1]; // b64 load
        v2f bb; bb.x = W[k0 + khalf];    bb.y = W[k0 + khalf + 1];
        acc = __builtin_amdgcn_wmma_f32_16x16x4_f32(
            /*neg_a=*/false, a, /*neg_b=*/false, bb,
            /*c_mod=*/(short)0, acc, /*reuse_a=*/false, /*reuse_b=*/false);
    }
    // D layout: VGPR r, lanes 0-15 -> M=r (N=lane); lanes 16-31 -> M=r+8.
    // Column 0 lives in lane 0 (rows 0-7) and lane 16 (rows 8-15).
    if ((lane & 15) == 0) {
        int mbase = row_base + ((lane >> 4) << 3);
#pragma unroll
        for (int r = 0; r < 8; ++r) h[mbase + r] = acc[r];
    }
}

// ---- dinv = rsqrt(deg); seed score with self-loop term dinv^2 * h --------
__global__ void k_dinv(const float* __restrict__ deg, const float* __restrict__ h,
                       float* __restrict__ dinv, float* __restrict__ score) {
    int i = blockIdx.x * blockDim.x + threadIdx.x;
    if (i < NTOT) {
        float d = deg[i];
        float r = (d > 0.0f) ? rsqrtf(d) : 0.0f;
        dinv[i] = r;
        score[i] = r * r * h[i];
    }
}

// ---- score[c] += dinv[r] * ew * dinv[c] * h[r] over edges ----------------
__global__ void k_score_edges(const int* __restrict__ ei, const float* __restrict__ ew,
                              const float* __restrict__ dinv, const float* __restrict__ h,
                              float* __restrict__ score) {
    int e = blockIdx.x * blockDim.x + threadIdx.x;
    if (e < NE) {
        int r = ei[e];
        int c = ei[NE + e];
        atomicAdd(&score[c], dinv[r] * ew[e] * dinv[c] * h[r]);
    }
}

// ---- score = tanh(score + b) ---------------------------------------------
__global__ void k_tanh(float* __restrict__ score, const float* __restrict__ b) {
    int i = blockIdx.x * blockDim.x + threadIdx.x;
    if (i < NTOT) score[i] = tanhf(score[i] + b[0]);
}

// ---- per-graph top-K with lax.top_k ordering & tie-breaking --------------
__global__ void k_topk(const float* __restrict__ score, float* __restrict__ mask,
                       float* __restrict__ perm_out) {
    __shared__ float s[NN];
    int g = blockIdx.x;
    int t = threadIdx.x;            // 0..511
    int gi = g * NN + t;
    float v = score[gi];
    s[t] = v;
    __syncthreads();
    int rank = 0;
    for (int j = 0; j < NN; ++j) {
        float sj = s[j];
        rank += (sj > v) || (sj == v && j < t);
    }
    if (rank < KK) {
        mask[gi] = 1.0f;
        perm_out[g * KK + rank] = (float)gi;   // descending-score order
    }
}

// ---- x2 = x * (score * mask), float4 vectorized --------------------------
__global__ void k_x2(const float* __restrict__ x, const float* __restrict__ score,
                     const float* __restrict__ mask, float* __restrict__ out) {
    size_t idx = (size_t)blockIdx.x * blockDim.x + threadIdx.x;  // one float4 each
    size_t node = idx / (DIM / 4);
    int c4 = (int)(idx % (DIM / 4));
    float sc = score[node] * mask[node];
    float4 vv = ((const float4*)(x + node * DIM))[c4];
    float4 o = make_float4(vv.x * sc, vv.y * sc, vv.z * sc, vv.w * sc);
    ((float4*)(out + node * DIM))[c4] = o;
}

// ---- edge_weight2 + edge_index passthrough -------------------------------
__global__ void k_edges_out(const int* __restrict__ ei, const float* __restrict__ ew,
                            const float* __restrict__ mask,
                            float* __restrict__ out_ei, float* __restrict__ out_ew) {
    int e = blockIdx.x * blockDim.x + threadIdx.x;
    if (e < NE) {
        int r = ei[e];
        int c = ei[NE + e];
        out_ei[e]      = (float)r;
        out_ei[NE + e] = (float)c;
        float mm = mask[r] + mask[c];
        out_ew[e] = (mm == 2.0f) ? ew[e] : 0.0f;
    }
}

// ---- batch passthrough ---------------------------------------------------
__global__ void k_batch_out(const int* __restrict__ batch, float* __restrict__ out_b) {
    int i = blockIdx.x * blockDim.x + threadIdx.x;
    if (i < NTOT) out_b[i] = (float)batch[i];
}

extern "C" void kernel_launch(void* const* d_in, const int* in_sizes, int n_in,
                              void* d_out, int out_size, void* d_ws, size_t ws_size,
                              hipStream_t stream) {
    (void)in_sizes; (void)n_in; (void)out_size; (void)ws_size;
    const float* x     = (const float*)d_in[0];
    const int*   ei    = (const int*)d_in[1];
    const float* ew    = (const float*)d_in[2];
    const int*   batch = (const int*)d_in[3];
    const float* W     = (const float*)d_in[4];
    const float* b     = (const float*)d_in[5];

    float* out      = (float*)d_out;
    float* out_x2   = out;                               // NTOT*DIM
    float* out_ei   = out_x2 + (size_t)NTOT * DIM;       // 2*NE
    float* out_ew   = out_ei + 2 * (size_t)NE;           // NE
    float* out_bat  = out_ew + (size_t)NE;               // NTOT
    float* out_perm = out_bat + NTOT;                    // NB*KK

    float* ws    = (float*)d_ws;
    float* h     = ws;
    float* deg   = ws + (size_t)NTOT;
    float* dinv  = ws + 2 * (size_t)NTOT;
    float* score = ws + 3 * (size_t)NTOT;
    float* mask  = ws + 4 * (size_t)NTOT;

    k_init<<<NTOT / 256, 256, 0, stream>>>(deg, mask);
    k_deg<<<NE / 256, 256, 0, stream>>>(ei, ew, deg);
    k_h_wmma<<<(NTOT / 16) * 32 / 256, 256, 0, stream>>>(x, W, h);
    k_dinv<<<NTOT / 256, 256, 0, stream>>>(deg, h, dinv, score);
    k_score_edges<<<NE / 256, 256, 0, stream>>>(ei, ew, dinv, h, score);
    k_tanh<<<NTOT / 256, 256, 0, stream>>>(score, b);
    k_topk<<<NB, NN, 0, stream>>>(score, mask, out_perm);
    k_x2<<<(size_t)NTOT * (DIM / 4) / 256, 256, 0, stream>>>(x, score, mask, out_x2);
    k_edges_out<<<NE / 256, 256, 0, stream>>>(ei, ew, mask, out_ei, out_ew);
    k_batch_out<<<NTOT / 256, 256, 0, stream>>>(batch, out_bat);
}